// SpatialEncoder_28114855920014
// MI455X (gfx1250) — compile-verified
//
#include <hip/hip_runtime.h>
#include <hip/hip_bf16.h>
#include <math.h>

// Problem constants (match reference)
#define NPTS 16384
#define H    256
#define HD   128     // H/2
#define KTOP 7       // k+1 smallest (includes self)

typedef float v2f __attribute__((ext_vector_type(2)));
typedef float v8f __attribute__((ext_vector_type(8)));

#if defined(__has_builtin)
#if __has_builtin(__builtin_amdgcn_tensor_load_to_lds)
#define HAVE_TDM 1
#endif
#endif

__device__ __forceinline__ float silu_f(float x) {
    return x / (1.0f + __expf(-x));
}

// ---------------------------------------------------------------------------
// K1: h = silu(x @ W1 + b1)   (N x H)   one thread per output element
// ---------------------------------------------------------------------------
__global__ __launch_bounds__(256) void coord_l1_kernel(
    const float* __restrict__ coords, const float* __restrict__ W1,
    const float* __restrict__ b1, float* __restrict__ h)
{
    int idx = blockIdx.x * 256 + threadIdx.x;           // [0, N*H)
    int n = idx >> 8;
    int j = idx & (H - 1);
    float x0 = coords[2 * n + 0];
    float x1 = coords[2 * n + 1];
    float pre = fmaf(x0, W1[j], fmaf(x1, W1[H + j], b1[j]));
    h[idx] = silu_f(pre);
}

// ---------------------------------------------------------------------------
// K2: per-row mean of 6 nearest-neighbor distances (excluding self).
// One thread per row; coords streamed through LDS in 2048-point tiles.
// Top-7 squared distances kept sorted ascending in registers.
// ---------------------------------------------------------------------------
#define KNN_TILE 2048

__global__ __launch_bounds__(256) void knn_mean_kernel(
    const float* __restrict__ coords, float* __restrict__ md)
{
    __shared__ float sx[KNN_TILE];
    __shared__ float sy[KNN_TILE];

    const int i = blockIdx.x * 256 + threadIdx.x;
    const float xi0 = coords[2 * i + 0];
    const float xi1 = coords[2 * i + 1];

    float m[KTOP];
#pragma unroll
    for (int t = 0; t < KTOP; ++t) m[t] = 3.4e38f;

    for (int t0 = 0; t0 < NPTS; t0 += KNN_TILE) {
        for (int u = threadIdx.x; u < KNN_TILE; u += 256) {
            sx[u] = coords[2 * (t0 + u) + 0];
            sy[u] = coords[2 * (t0 + u) + 1];
        }
        __syncthreads();
#pragma unroll 4
        for (int u = 0; u < KNN_TILE; ++u) {
            float dx = sx[u] - xi0;
            float dy = sy[u] - xi1;
            float d2 = fmaf(dx, dx, dy * dy);
            if (d2 < m[KTOP - 1]) {          // rare: ~7*ln(N) hits per row
                float x = d2;
#pragma unroll
                for (int t = 0; t < KTOP; ++t) {   // sorted insert (min/max chain)
                    float lo = fminf(m[t], x);
                    x        = fmaxf(m[t], x);
                    m[t] = lo;
                }
            }
        }
        __syncthreads();
    }

    // m[0] ~ self (0); mean sqrt of the next 6 (matches topk[:,1:])
    float s = 0.0f;
#pragma unroll
    for (int t = 1; t < KTOP; ++t)
        s += sqrtf(fmaxf(m[t], 1e-12f));
    md[i] = s * (1.0f / 6.0f);
}

// ---------------------------------------------------------------------------
// K3: g = silu(md @ Wd1 + bd1)   (N x HD)
// ---------------------------------------------------------------------------
__global__ __launch_bounds__(256) void dist_l1_kernel(
    const float* __restrict__ md, const float* __restrict__ Wd1,
    const float* __restrict__ bd1, float* __restrict__ g)
{
    int idx = blockIdx.x * 256 + threadIdx.x;           // [0, N*HD)
    int n = idx >> 7;
    int j = idx & (HD - 1);
    float pre = fmaf(md[n], Wd1[j], bd1[j]);
    g[idx] = silu_f(pre);
}

// ---------------------------------------------------------------------------
// TDM 2D tile load: global (rows x dim0 f32, row stride = stride elems)
// -> LDS at lds_off, with 1-DWORD pad inserted every (2<<pad_code) DWORDs
// (pad_code 7 => every 256 DW => LDS row stride 257; 6 => 129).
// D# layout per CDNA5 ISA ch.8. Issued once by one wave; TENSORcnt tracked.
// clang-23 toolchain: 6-arg builtin (extra int32x8 before cache policy).
// ---------------------------------------------------------------------------
#ifdef HAVE_TDM
typedef unsigned int u32x4 __attribute__((ext_vector_type(4)));
typedef int          i32x8 __attribute__((ext_vector_type(8)));
typedef int          i32x4 __attribute__((ext_vector_type(4)));

__device__ __forceinline__ void tdm_load_tile_2d(
    unsigned lds_off, const float* gptr,
    unsigned dim0, unsigned rows, unsigned stride,
    unsigned pad_code, unsigned tensor_rows)
{
    unsigned long long ga = (unsigned long long)(const void*)gptr;
    u32x4 g0;
    g0[0] = 1u;                                        // count=1, user descriptor
    g0[1] = lds_off;                                   // lds_addr (bytes)
    g0[2] = (unsigned)(ga & 0xFFFFFFFFull);            // global_addr[31:0]
    g0[3] = (unsigned)((ga >> 32) & 0x1FFFFFFull)      // global_addr[56:32]
          | (2u << 30);                                // type = 2 ("image")
    i32x8 g1;
    g1[0] = (int)((2u << 16)                           // data_size = 4B
          | (1u << 20)                                 // pad_enable
          | (pad_code << 22));                         // pad_interval; pad_amount=0 (1 DW)
    g1[1] = (int)((dim0 & 0xFFFFu) << 16);             // tensor_dim0[15:0]
    g1[2] = (int)(((dim0 >> 16) & 0xFFFFu)             // tensor_dim0[31:16]
          | ((tensor_rows & 0xFFFFu) << 16));          // tensor_dim1[15:0]
    g1[3] = (int)(((tensor_rows >> 16) & 0xFFFFu)      // tensor_dim1[31:16]
          | ((dim0 & 0xFFFFu) << 16));                 // tile_dim0
    g1[4] = (int)(rows & 0xFFFFu);                     // tile_dim1; tile_dim2 = 0
    g1[5] = (int)stride;                               // tensor_dim0_stride[31:0]
    g1[6] = 0;                                         // stride[47:32], dim1_stride lo
    g1[7] = 0;                                         // dim1_stride hi
    i32x4 z4 = {0, 0, 0, 0};                           // groups 2/3 unused (2D tile)
    i32x8 z8 = {0, 0, 0, 0, 0, 0, 0, 0};
    __builtin_amdgcn_tensor_load_to_lds(g0, g1, z4, z4, z8, 0);
}
#endif

// ---------------------------------------------------------------------------
// K4: out = h @ W2 + b2 + g @ Wd2 + bd2, fused via V_WMMA_F32_16X16X4_F32.
// Block = 512 threads = 16 waves; block handles 16 output rows, wave w
// handles the 16-column tile starting at 16*w. A tiles (h, g) staged in LDS
// by the Tensor Data Mover with hardware padding (row stride 257 / 129
// floats -> lanes hit distinct banks).
// ---------------------------------------------------------------------------
#define LH_STRIDE (H + 1)     // 257
#define LG_STRIDE (HD + 1)    // 129

__global__ __launch_bounds__(512) void fused_gemm_kernel(
    const float* __restrict__ h,   const float* __restrict__ g,
    const float* __restrict__ W2,  const float* __restrict__ b2,
    const float* __restrict__ Wd2, const float* __restrict__ bd2,
    float* __restrict__ out)
{
    __shared__ float lh[16 * LH_STRIDE];
    __shared__ float lg[16 * LG_STRIDE];

    const int tid  = threadIdx.x;
    const int row0 = blockIdx.x * 16;

#ifdef HAVE_TDM
    if (tid < 32) {                      // wave 0 issues both TDM descriptors
        tdm_load_tile_2d((unsigned)(size_t)lh, h + (size_t)row0 * H,
                         H,  16, H,  7, (unsigned)(NPTS - row0));
        tdm_load_tile_2d((unsigned)(size_t)lg, g + (size_t)row0 * HD,
                         HD, 16, HD, 6, (unsigned)(NPTS - row0));
        __builtin_amdgcn_s_wait_tensorcnt(0);
    }
#else
    for (int idx = tid; idx < 16 * H; idx += 512) {
        int r = idx >> 8, c = idx & (H - 1);
        lh[r * LH_STRIDE + c] = h[(size_t)(row0 + r) * H + c];
    }
    for (int idx = tid; idx < 16 * HD; idx += 512) {
        int r = idx >> 7, c = idx & (HD - 1);
        lg[r * LG_STRIDE + c] = g[(size_t)(row0 + r) * HD + c];
    }
#endif
    __syncthreads();

    const int wave    = tid >> 5;
    const int lane    = tid & 31;
    const int half    = lane >> 4;    // 0: K/M rows 0..; 1: K+2 / M+8
    const int l16     = lane & 15;
    const int colBase = wave * 16;
    const int col     = colBase + l16;
    const int kb      = half * 2;

    // C init: biases broadcast down the column
    float bias = b2[col] + bd2[col];
    v8f c;
#pragma unroll
    for (int v = 0; v < 8; ++v) c[v] = bias;

    // GEMM 1: (16 x 256) @ (256 x 16)  -> 64 WMMA ops
#pragma unroll 4
    for (int k0 = 0; k0 < H; k0 += 4) {
        v2f a, b;
        a.x = lh[l16 * LH_STRIDE + k0 + kb];
        a.y = lh[l16 * LH_STRIDE + k0 + kb + 1];
        b.x = W2[(size_t)(k0 + kb)     * H + col];
        b.y = W2[(size_t)(k0 + kb + 1) * H + col];
        c = __builtin_amdgcn_wmma_f32_16x16x4_f32(
                false, a, false, b, (short)0, c, false, false);
    }
    // GEMM 2: (16 x 128) @ (128 x 16)  -> 32 WMMA ops
#pragma unroll 4
    for (int k0 = 0; k0 < HD; k0 += 4) {
        v2f a, b;
        a.x = lg[l16 * LG_STRIDE + k0 + kb];
        a.y = lg[l16 * LG_STRIDE + k0 + kb + 1];
        b.x = Wd2[(size_t)(k0 + kb)     * H + col];
        b.y = Wd2[(size_t)(k0 + kb + 1) * H + col];
        c = __builtin_amdgcn_wmma_f32_16x16x4_f32(
                false, a, false, b, (short)0, c, false, false);
    }

    // D layout: VGPR v holds row (v + 8*half), column l16
#pragma unroll
    for (int v = 0; v < 8; ++v) {
        int mrow = v + half * 8;
        out[(size_t)(row0 + mrow) * H + col] = c[v];
    }
}

// ---------------------------------------------------------------------------
// Launch: h and g scratch live in d_ws (needs ~25.3 MB).
// ---------------------------------------------------------------------------
extern "C" void kernel_launch(void* const* d_in, const int* in_sizes, int n_in,
                              void* d_out, int out_size, void* d_ws, size_t ws_size,
                              hipStream_t stream) {
    const float* coords = (const float*)d_in[0];
    const float* W1     = (const float*)d_in[1];
    const float* b1     = (const float*)d_in[2];
    const float* W2     = (const float*)d_in[3];
    const float* b2     = (const float*)d_in[4];
    const float* Wd1    = (const float*)d_in[5];
    const float* bd1    = (const float*)d_in[6];
    const float* Wd2    = (const float*)d_in[7];
    const float* bd2    = (const float*)d_in[8];
    // d_in[9] is k (==6); KTOP hardcoded to k+1=7.

    float* ws = (float*)d_ws;
    float* h  = ws;                                    // N*H floats
    float* g  = h + (size_t)NPTS * H;                  // N*HD floats
    float* md = g + (size_t)NPTS * HD;                 // N floats
    float* out = (float*)d_out;

    coord_l1_kernel<<<(NPTS * H) / 256, 256, 0, stream>>>(coords, W1, b1, h);
    knn_mean_kernel<<<NPTS / 256, 256, 0, stream>>>(coords, md);
    dist_l1_kernel<<<(NPTS * HD) / 256, 256, 0, stream>>>(md, Wd1, bd1, g);
    fused_gemm_kernel<<<NPTS / 16, 512, 0, stream>>>(h, g, W2, b2, Wd2, bd2, out);
}